// Task_FSQ_Transformer_14834817040791
// MI455X (gfx1250) — compile-verified
//
#include <hip/hip_runtime.h>
#include <hip/hip_bf16.h>
#include <math.h>

typedef __attribute__((ext_vector_type(2))) float v2f;
typedef __attribute__((ext_vector_type(8))) float v8f;

#define B_    8
#define SMAX  1024
#define SLAT  256
#define DM    1024
#define LD    4
#define CBN   2025          // 9*9*5*5
#define ZIDX  1012          // codebook index of all-zero latent ((4*9+4)*5+2)*5+2

// output layout (floats): x_rec | z_e | z_q | usage | unique | percent_masked
#define OFF_ZE    (B_*SMAX*DM)              // 8388608
#define OFF_ZQ    (OFF_ZE + B_*SMAX*LD)     // 8421376
#define OFF_SCAL  (OFF_ZQ + B_*SMAX*LD)     // 8454144

// ---------------------------------------------------------------------------
// Init: zero codebook-usage flags (flag[1012]=1: all masked rows hit it) and
// zero z_e / z_q for the masked rows (s >= 256).
// ---------------------------------------------------------------------------
__global__ void fsq_init(float* __restrict__ out, int* __restrict__ flags) {
  const int t = blockIdx.x * blockDim.x + threadIdx.x;
  if (t < CBN) flags[t] = (t == ZIDX) ? 1 : 0;
  const int n = B_ * (SMAX - SLAT) * LD;     // 24576
  if (t < n) {
    const int row = t / LD;
    const int b = row / (SMAX - SLAT);
    const int s = SLAT + (row % (SMAX - SLAT));
    const int rg = b * SMAX + s;
    const int j = t % LD;
    out[OFF_ZE + rg * LD + j] = 0.0f;
    out[OFF_ZQ + rg * LD + j] = 0.0f;
  }
}

// ---------------------------------------------------------------------------
// Encode + FSQ quantize. One wave32 per active row (2048 rows).
// Each lane streams float4's of x (512B/wave/iter coalesced), accumulates the
// 4 latent dots, butterfly-reduces across the wave, then lane 0 does
// tanh/round/index and marks the usage flag.
// ---------------------------------------------------------------------------
__global__ void fsq_encode(const float* __restrict__ x,
                           const float* __restrict__ W_enc,
                           const float* __restrict__ b_enc,
                           float* __restrict__ out,
                           int* __restrict__ flags) {
  const int wave = (blockIdx.x * blockDim.x + threadIdx.x) >> 5;
  const int lane = threadIdx.x & 31;
  if (wave >= B_ * SLAT) return;
  const int b = wave / SLAT;
  const int s = wave % SLAT;
  const int rg = b * SMAX + s;

  const float4* __restrict__ xr = (const float4*)(x + (size_t)rg * DM);
  const float4* __restrict__ We = (const float4*)W_enc;   // row k = W_enc[k][0..3]

  float a0 = 0.f, a1 = 0.f, a2 = 0.f, a3 = 0.f;
#pragma unroll
  for (int i = 0; i < DM / 128; ++i) {
    const int kq = lane + 32 * i;        // float4 index within the row
    const float4 xv = xr[kq];
    const int kb = kq * 4;
    const float4 w0 = We[kb + 0];
    const float4 w1 = We[kb + 1];
    const float4 w2 = We[kb + 2];
    const float4 w3 = We[kb + 3];
    a0 += xv.x * w0.x + xv.y * w1.x + xv.z * w2.x + xv.w * w3.x;
    a1 += xv.x * w0.y + xv.y * w1.y + xv.z * w2.y + xv.w * w3.y;
    a2 += xv.x * w0.z + xv.y * w1.z + xv.z * w2.z + xv.w * w3.z;
    a3 += xv.x * w0.w + xv.y * w1.w + xv.z * w2.w + xv.w * w3.w;
  }
#pragma unroll
  for (int off = 16; off > 0; off >>= 1) {
    a0 += __shfl_xor(a0, off, 32);
    a1 += __shfl_xor(a1, off, 32);
    a2 += __shfl_xor(a2, off, 32);
    a3 += __shfl_xor(a3, off, 32);
  }
  if (lane == 0) {
    const float h0 = a0 + b_enc[0];
    const float h1 = a1 + b_enc[1];
    const float h2 = a2 + b_enc[2];
    const float h3 = a3 + b_enc[3];
    // halfL = [4,4,2,2]; rintf == round-half-even == jnp.round
    const float q0 = rintf(4.0f * tanhf(h0));
    const float q1 = rintf(4.0f * tanhf(h1));
    const float q2 = rintf(2.0f * tanhf(h2));
    const float q3 = rintf(2.0f * tanhf(h3));
    float* ze = out + OFF_ZE + rg * LD;
    float* zq = out + OFF_ZQ + rg * LD;
    ze[0] = h0; ze[1] = h1; ze[2] = h2; ze[3] = h3;
    zq[0] = q0; zq[1] = q1; zq[2] = q2; zq[3] = q3;
    const int idx = ((((int)q0 + 4) * 9 + ((int)q1 + 4)) * 5 + ((int)q2 + 2)) * 5
                    + ((int)q3 + 2);
    flags[idx] = 1;                       // racing stores of 1 are benign
  }
}

// ---------------------------------------------------------------------------
// Usage statistics: U = #used codebook entries.
// usage = (U + (C-U)*e^-1)/C ; unique = U ; percent_masked = 0
// ---------------------------------------------------------------------------
__global__ void fsq_stats(const int* __restrict__ flags, float* __restrict__ out) {
  __shared__ int sh[256];
  int acc = 0;
  for (int i = threadIdx.x; i < CBN; i += 256) acc += flags[i];
  sh[threadIdx.x] = acc;
  __syncthreads();
#pragma unroll
  for (int o = 128; o > 0; o >>= 1) {
    if ((int)threadIdx.x < o) sh[threadIdx.x] += sh[threadIdx.x + o];
    __syncthreads();
  }
  if (threadIdx.x == 0) {
    const float U = (float)sh[0];
    out[OFF_SCAL + 0] = (U + ((float)CBN - U) * 0.36787944117144233f) / (float)CBN;
    out[OFF_SCAL + 1] = U;
    out[OFF_SCAL + 2] = 0.0f;
  }
}

// ---------------------------------------------------------------------------
// Decode GEMM via V_WMMA_F32_16X16X4_F32:  x_rec[8192,1024] = z@W_dec + b_dec
// K=4 exactly matches the WMMA shape -> one wmma per 16x16 tile.
// Block = 8 waves; each wave owns one M-tile's A operand and sweeps 8 N-tiles
// with a constant-trip fully-unrolled loop so all W_dec/b_dec loads and the
// 64 output stores use immediate offsets off a single per-lane base.
// A layout (ISA 7.12.2, 32-bit A 16x4): lanes 0-15 carry K={0,1}, lanes 16-31
// carry K={2,3}, M = lane&15. C/D: VGPR p = rows {p, p+8} across lane halves.
// ---------------------------------------------------------------------------
__global__ void fsq_decode_wmma(const float* __restrict__ zq,
                                const float* __restrict__ W_dec,   // [4,1024]
                                const float* __restrict__ b_dec,
                                float* __restrict__ xrec) {
  const int lane = threadIdx.x & 31;
  const int wv   = threadIdx.x >> 5;        // 0..7
  const int tm   = blockIdx.x;              // 0..511 M-tiles
  const int half = lane >> 4;               // 0|1
  const int l15  = lane & 15;
  const int kh   = half * 2;                // K base {0,2}

  // A tile: loaded once per wave, reused for all 8 WMMAs
  const int arow = tm * 16 + l15;
  v2f A;
  A.x = zq[arow * 4 + kh + 0];
  A.y = zq[arow * 4 + kh + 1];

  // Per-lane base pointers; per-iteration displacement is it*128 floats (512B)
  const int n0 = wv * 16 + l15;
  const float* __restrict__ Wd0 = W_dec + (kh + 0) * DM + n0;
  const float* __restrict__ Wd1 = W_dec + (kh + 1) * DM + n0;
  const float* __restrict__ bd  = b_dec + n0;
  float* __restrict__ outp = xrec + (size_t)(tm * 16 + half * 8) * DM + n0;

#pragma unroll
  for (int it = 0; it < 8; ++it) {
    v2f Bv;
    Bv.x = Wd0[it * 128];
    Bv.y = Wd1[it * 128];
    v8f C = {};
    C = __builtin_amdgcn_wmma_f32_16x16x4_f32(
        /*neg_a=*/false, A, /*neg_b=*/false, Bv,
        /*c_mod=*/(short)0, C, /*reuse_a=*/false, /*reuse_b=*/false);
    const float bias = bd[it * 128];
#pragma unroll
    for (int p = 0; p < 8; ++p) {
      // byte offset = p*4096 + it*512, fits the 24-bit immediate
      outp[(size_t)p * DM + it * 128] = C[p] + bias;
    }
  }
}

extern "C" void kernel_launch(void* const* d_in, const int* in_sizes, int n_in,
                              void* d_out, int out_size, void* d_ws, size_t ws_size,
                              hipStream_t stream) {
  const float* x     = (const float*)d_in[0];
  const float* W_enc = (const float*)d_in[1];
  const float* b_enc = (const float*)d_in[2];
  const float* W_dec = (const float*)d_in[3];
  const float* b_dec = (const float*)d_in[4];
  // d_in[5] (codebook) unused: index computed analytically (lexicographic grid)
  float* out  = (float*)d_out;
  int* flags  = (int*)d_ws;                 // 2025 ints of scratch

  fsq_init<<<96, 256, 0, stream>>>(out, flags);
  fsq_encode<<<(B_ * SLAT * 32) / 256, 256, 0, stream>>>(x, W_enc, b_enc, out, flags);
  fsq_stats<<<1, 256, 0, stream>>>(flags, out);
  fsq_decode_wmma<<<(B_ * SMAX) / 16, 256, 0, stream>>>(out + OFF_ZQ, W_dec, b_dec, out);
}